// TopKTiledSoftmax_56307021251147
// MI455X (gfx1250) — compile-verified
//
#include <hip/hip_runtime.h>
#include <hip/hip_bf16.h>
#include <math.h>

// Problem constants (match reference)
#define TOKENS      4096
#define IN_DIM      256
#define VOCAB       32000
#define TOPK        128
#define TILE_FACTOR 16
#define TILE_SZ     (VOCAB / TILE_FACTOR)   // 2000
#define PAD_N       2048                    // tile padded to pow2 for bitonic
#define NEG_BIG     (-1e30f)

typedef float v2f __attribute__((ext_vector_type(2)));
typedef float v8f __attribute__((ext_vector_type(8)));

#define WMMA_F32(A, B, C) \
    __builtin_amdgcn_wmma_f32_16x16x4_f32(false, (A), false, (B), (short)0, (C), false, false)

// ---------------------------------------------------------------------------
// Kernel 1: logits = x @ W^T via V_WMMA_F32_16X16X4_F32 (fp32 WMMA, wave32).
// Register-blocked: each wave owns a 32x64 macro-tile (2 M-tiles x 4 N-tiles,
// 8 fp32 accumulators = 64 VGPRs). Per k-step: 6 loads feed 8 WMMAs.
// Block = 256 threads = 8 waves arranged 2(M) x 4(N) -> block tile 64x256.
// grid = (VOCAB/256, TOKENS/64) = (125, 64).
// ---------------------------------------------------------------------------
__global__ __launch_bounds__(256)
void gemm_logits_wmma(const float* __restrict__ x,
                      const float* __restrict__ w,
                      float* __restrict__ logits) {
    const int lane  = threadIdx.x & 31;
    const int wave  = threadIdx.x >> 5;    // 0..7
    const int waveM = wave >> 2;           // 0..1
    const int waveN = wave & 3;            // 0..3

    const int row0 = blockIdx.y * 64  + waveM * 32;   // token rows (2 tiles)
    const int col0 = blockIdx.x * 256 + waveN * 64;   // vocab cols (4 tiles)

    // Fragment addressing per CDNA5 ISA 7.12.2 (wave32):
    //  A (16x4 f32): lanes 0-15 -> M=lane, K={0,1}; lanes 16-31 -> K={2,3}
    //  B (4x16 f32): lanes 0-15 -> N=lane, K striped the same way
    const int mn    = lane & 15;
    const int khalf = lane >> 4;           // 0/1 -> K offset 0/2

    const float* aBase = x + (size_t)(row0 + mn) * IN_DIM + 2 * khalf;
    const float* bBase = w + (size_t)(col0 + mn) * IN_DIM + 2 * khalf;

    v8f c00 = {}, c01 = {}, c02 = {}, c03 = {};
    v8f c10 = {}, c11 = {}, c12 = {}, c13 = {};

#pragma unroll 2
    for (int kk = 0; kk < IN_DIM / 4; ++kk) {
        const int ko = kk * 4;
        // 2 A fragments (M tiles 16 rows apart), 4 B fragments (N tiles)
        v2f A0 = *(const v2f*)(aBase + ko);
        v2f A1 = *(const v2f*)(aBase + 16 * IN_DIM + ko);
        v2f B0 = *(const v2f*)(bBase + ko);
        v2f B1 = *(const v2f*)(bBase + 16 * IN_DIM + ko);
        v2f B2 = *(const v2f*)(bBase + 32 * IN_DIM + ko);
        v2f B3 = *(const v2f*)(bBase + 48 * IN_DIM + ko);

        c00 = WMMA_F32(A0, B0, c00);
        c01 = WMMA_F32(A0, B1, c01);
        c02 = WMMA_F32(A0, B2, c02);
        c03 = WMMA_F32(A0, B3, c03);
        c10 = WMMA_F32(A1, B0, c10);
        c11 = WMMA_F32(A1, B1, c11);
        c12 = WMMA_F32(A1, B2, c12);
        c13 = WMMA_F32(A1, B3, c13);
    }

    // C/D layout: VGPR r holds M = r + 8*khalf, N = lane&15
    v8f acc[2][4] = {{c00, c01, c02, c03}, {c10, c11, c12, c13}};
#pragma unroll
    for (int tm = 0; tm < 2; ++tm) {
#pragma unroll
        for (int r = 0; r < 8; ++r) {
            const size_t row = row0 + tm * 16 + r + 8 * khalf;
            float* dst = logits + row * VOCAB + col0 + mn;
#pragma unroll
            for (int tn = 0; tn < 4; ++tn)
                dst[tn * 16] = acc[tm][tn][r];
        }
    }
}

// ---------------------------------------------------------------------------
// Shared bitonic sort: descending by value, ascending index tiebreak.
// n must be a power of two; arrays live in LDS.
// ---------------------------------------------------------------------------
__device__ __forceinline__ void bitonic_desc(float* v, int* idx, int n,
                                             int tid, int nthreads) {
    for (int k = 2; k <= n; k <<= 1) {
        for (int j = k >> 1; j > 0; j >>= 1) {
            for (int i = tid; i < n; i += nthreads) {
                int l = i ^ j;
                if (l > i) {
                    float vi = v[i], vl = v[l];
                    int   ii = idx[i], il = idx[l];
                    bool iBetter = (vi > vl) || (vi == vl && ii < il);
                    bool desc    = ((i & k) == 0);
                    if (desc ? !iBetter : iBetter) {
                        v[i] = vl; v[l] = vi;
                        idx[i] = il; idx[l] = ii;
                    }
                }
            }
            __syncthreads();
        }
    }
}

// ---------------------------------------------------------------------------
// Kernel 2: per-(token, tile) top-128 of 2000 logits -> workspace.
// grid = (TILE_FACTOR, TOKENS), block = 256.
// ---------------------------------------------------------------------------
__global__ __launch_bounds__(256)
void topk_per_tile(const float* __restrict__ logits,
                   float* __restrict__ wsVals,
                   int*   __restrict__ wsIdx) {
    __shared__ float sv[PAD_N];
    __shared__ int   si[PAD_N];

    const int tid   = threadIdx.x;
    const int tile  = blockIdx.x;
    const int token = blockIdx.y;

    const float* seg = logits + (size_t)token * VOCAB + tile * TILE_SZ;

    for (int i = tid; i < PAD_N; i += 256) {
        sv[i] = (i < TILE_SZ) ? seg[i] : NEG_BIG * 2.0f;  // pad below sentinel
        si[i] = i;                                        // local index in tile
    }
    __syncthreads();

    bitonic_desc(sv, si, PAD_N, tid, 256);

    // write top-128 (value, GLOBAL vocab index)
    float* ov = wsVals + ((size_t)token * TILE_FACTOR + tile) * TOPK;
    int*   oi = wsIdx  + ((size_t)token * TILE_FACTOR + tile) * TOPK;
    if (tid < TOPK) {
        ov[tid] = sv[tid];
        oi[tid] = tile * TILE_SZ + si[tid];
    }
}

// ---------------------------------------------------------------------------
// Kernel 3: per-token merge (2048 -> 128), target logit, sparse softmax,
// row zeroing + scatter into d_out. grid = TOKENS, block = 256.
// ---------------------------------------------------------------------------
__global__ __launch_bounds__(256)
void merge_softmax_scatter(const float* __restrict__ x,
                           const float* __restrict__ w,
                           const int*   __restrict__ target,
                           const float* __restrict__ wsVals,
                           const int*   __restrict__ wsIdx,
                           float* __restrict__ out) {
    __shared__ float sv[PAD_N];
    __shared__ int   si[PAD_N];
    __shared__ float red[256];
    __shared__ float ev[TOPK + 1];
    __shared__ int   tpos;
    __shared__ float sM, sS, sTgt;

    const int tid   = threadIdx.x;
    const int token = blockIdx.x;
    const int tgt   = target[token];

    // load 16*128 candidates
    const float* cv = wsVals + (size_t)token * (TILE_FACTOR * TOPK);
    const int*   ci = wsIdx  + (size_t)token * (TILE_FACTOR * TOPK);
    for (int i = tid; i < PAD_N; i += 256) {
        sv[i] = cv[i];
        si[i] = ci[i];
    }
    if (tid == 0) tpos = -1;
    __syncthreads();

    bitonic_desc(sv, si, PAD_N, tid, 256);  // top-128 now in [0,128)

    // target logit: dot(x[token], w[tgt]) over 256 dims (one elem/thread)
    red[tid] = x[(size_t)token * IN_DIM + tid] * w[(size_t)tgt * IN_DIM + tid];
    __syncthreads();
    for (int s = 128; s > 0; s >>= 1) {
        if (tid < s) red[tid] += red[tid + s];
        __syncthreads();
    }
    if (tid == 0) sTgt = red[0];

    // is target among top-128? (indices are unique -> at most one hit)
    if (tid < TOPK && si[tid] == tgt) tpos = tid;
    __syncthreads();

    const float tgtLogit = sTgt;
    const int   count    = (tpos < 0) ? TOPK + 1 : TOPK;

    // reference semantics: duplicate scatter-add => target entry is doubled
    if (tid < TOPK) ev[tid] = sv[tid] + ((tid == tpos) ? tgtLogit : 0.0f);
    if (tid == 0 && tpos < 0) ev[TOPK] = tgtLogit;
    __syncthreads();

    // max over entries
    float lm = -INFINITY;
    for (int i = tid; i < count; i += 256) lm = fmaxf(lm, ev[i]);
    red[tid] = lm;
    __syncthreads();
    for (int s = 128; s > 0; s >>= 1) {
        if (tid < s) red[tid] = fmaxf(red[tid], red[tid + s]);
        __syncthreads();
    }
    if (tid == 0) sM = red[0];
    __syncthreads();

    // sum of exp
    float ls = 0.0f;
    for (int i = tid; i < count; i += 256) ls += __expf(ev[i] - sM);
    red[tid] = ls;
    __syncthreads();
    for (int s = 128; s > 0; s >>= 1) {
        if (tid < s) red[tid] += red[tid + s];
        __syncthreads();
    }
    if (tid == 0) sS = red[0];
    __syncthreads();

    const float m = sM, invS = 1.0f / sS;

    // zero the full output row (32000 fp32 = 8000 float4, 16B aligned)
    float4* orow = (float4*)(out + (size_t)token * VOCAB);
    const float4 z = make_float4(0.f, 0.f, 0.f, 0.f);
    for (int i = tid; i < VOCAB / 4; i += 256) orow[i] = z;
    __threadfence_block();
    __syncthreads();

    // scatter softmax values
    float* row = out + (size_t)token * VOCAB;
    if (tid < TOPK)              row[si[tid]] = __expf(ev[tid] - m) * invS;
    if (tid == 0 && tpos < 0)    row[tgt]     = __expf(ev[TOPK] - m) * invS;
}

// ---------------------------------------------------------------------------
// Host-side launcher
// ---------------------------------------------------------------------------
extern "C" void kernel_launch(void* const* d_in, const int* in_sizes, int n_in,
                              void* d_out, int out_size, void* d_ws, size_t ws_size,
                              hipStream_t stream) {
    const float* x      = (const float*)d_in[0];   // [4096, 256]
    const float* w      = (const float*)d_in[1];   // [32000, 256]
    const int*   target = (const int*)d_in[2];     // [4096]
    float*       out    = (float*)d_out;           // [4096, 32000]

    // workspace layout: candidate vals then candidate indices (~67 MB)
    const size_t nCand = (size_t)TOKENS * TILE_FACTOR * TOPK;  // 8.39M
    float* wsVals = (float*)d_ws;
    int*   wsIdx  = (int*)((char*)d_ws + nCand * sizeof(float));

    // 1) logits into d_out (scratch reuse: same shape as final output)
    dim3 gGemm(VOCAB / 256, TOKENS / 64);
    gemm_logits_wmma<<<gGemm, 256, 0, stream>>>(x, w, out);

    // 2) per-tile top-128
    dim3 gTop(TILE_FACTOR, TOKENS);
    topk_per_tile<<<gTop, 256, 0, stream>>>(out, wsVals, wsIdx);

    // 3) merge + softmax + scatter (overwrites d_out rows completely)
    merge_softmax_scatter<<<TOKENS, 256, 0, stream>>>(x, w, target,
                                                      wsVals, wsIdx, out);
}